// DualAttention_34935263986206
// MI455X (gfx1250) — compile-verified
//
#include <hip/hip_runtime.h>
#include <stdint.h>

#define HH 224
#define WW 224
#define HW (HH * WW)          // 50176
#define NB 64
#define N4 (HW / 4)           // 12544 float4 per image
#define THREADS 256
#define ITERS (N4 / THREADS)  // 49
#define K2_BLOCKS_X 7
#define K2_CHUNKS 7           // 7 * 7 * 256 = 12544

// ---- CDNA5 async global->LDS path (ASYNCcnt) ------------------------------
// VDST = VGPR holding LDS byte offset; VADDR = 64-bit global address.
__device__ __forceinline__ void async_ld_b128(const void* lds_generic, const void* gaddr) {
    // Low 32 bits of a generic LDS pointer are the LDS byte offset (ISA 10.2).
    uint32_t lds_off = (uint32_t)(uintptr_t)lds_generic;
    asm volatile("global_load_async_to_lds_b128 %0, %1, off"
                 :: "v"(lds_off), "v"(gaddr)
                 : "memory");
}
__device__ __forceinline__ void wait_async_le2() {
    asm volatile("s_wait_asynccnt 2" ::: "memory");
}
__device__ __forceinline__ void wait_async_0() {
    asm volatile("s_wait_asynccnt 0" ::: "memory");
}

// ---- Pass 1: per-batch max/min of d = depth*object, + band params ---------
__global__ __launch_bounds__(THREADS) void reduce_params_kernel(
    const float* __restrict__ depth, const float* __restrict__ obj,
    const float* __restrict__ gaze,  const int* __restrict__ headp,
    float* __restrict__ params /* [NB][2] = {point_depth, fd_range} */) {

    __shared__ __align__(16) unsigned char smem[16384 + 64];
    float4* sd = (float4*)smem;             // [2][THREADS]
    float4* so = (float4*)(smem + 8192);    // [2][THREADS]
    float*  red = (float*)(smem + 16384);   // 8 max + 8 min

    const int b = blockIdx.x;
    const int t = threadIdx.x;
    const float4* dg = (const float4*)(depth + (size_t)b * HW);
    const float4* og = (const float4*)(obj   + (size_t)b * HW);

    // prologue: stage chunk 0
    async_ld_b128(&sd[t], &dg[t]);
    async_ld_b128(&so[t], &og[t]);

    float vmax = -3.402823466e38f, vmin = 3.402823466e38f;
    for (int k = 0; k < ITERS; ++k) {
        const int cb = (k & 1) * THREADS;
        if (k + 1 < ITERS) {
            const int nb = ((k + 1) & 1) * THREADS;
            const int g  = (k + 1) * THREADS + t;
            async_ld_b128(&sd[nb + t], &dg[g]);
            async_ld_b128(&so[nb + t], &og[g]);
            wait_async_le2();   // oldest 2 (chunk k) complete; in-order returns
        } else {
            wait_async_0();
        }
        float4 dv = sd[cb + t];
        float4 ov = so[cb + t];
        float p0 = dv.x * ov.x, p1 = dv.y * ov.y;
        float p2 = dv.z * ov.z, p3 = dv.w * ov.w;
        vmax = fmaxf(vmax, fmaxf(fmaxf(p0, p1), fmaxf(p2, p3)));
        vmin = fminf(vmin, fminf(fminf(p0, p1), fminf(p2, p3)));
    }

    // wave32 reduction, then cross-wave via LDS
    #pragma unroll
    for (int off = 16; off > 0; off >>= 1) {
        vmax = fmaxf(vmax, __shfl_xor(vmax, off, 32));
        vmin = fminf(vmin, __shfl_xor(vmin, off, 32));
    }
    const int wave = t >> 5, lane = t & 31;
    if (lane == 0) { red[wave] = vmax; red[8 + wave] = vmin; }
    __syncthreads();
    if (t == 0) {
        float m = red[0], n = red[8];
        #pragma unroll
        for (int i = 1; i < 8; ++i) { m = fmaxf(m, red[i]); n = fminf(n, red[8 + i]); }
        const float fr = (m - n) * (1.0f / 24.0f);
        const int hp0 = headp[2 * b], hp1 = headp[2 * b + 1];
        const size_t hidx = (size_t)b * HW + (size_t)hp0 * WW + (size_t)hp1;
        const float hd = depth[hidx] * obj[hidx];
        const float pd = hd + gaze[3 * b + 2] * 224.0f;
        params[2 * b]     = pd;
        params[2 * b + 1] = fr;
    }
}

// ---- Pass 2: cone mask * depth bands -> (NB, 3, H, W) ---------------------
__global__ __launch_bounds__(THREADS) void mask_band_kernel(
    const float* __restrict__ depth, const float* __restrict__ obj,
    const float* __restrict__ gaze,  const int* __restrict__ headp,
    const float* __restrict__ params, float* __restrict__ out) {

    __shared__ __align__(16) unsigned char smem[16384];
    float4* sd = (float4*)smem;             // [2][THREADS]
    float4* so = (float4*)(smem + 8192);    // [2][THREADS]

    const int b = blockIdx.y;               // uniform per block -> scalar loads
    const int t = threadIdx.x;
    const float4* dg = (const float4*)(depth + (size_t)b * HW);
    const float4* og = (const float4*)(obj   + (size_t)b * HW);
    float* out_b = out + (size_t)b * 3 * HW;

    const float pd = params[2 * b];
    const float fr = params[2 * b + 1];
    const float gx = gaze[3 * b], gy = gaze[3 * b + 1];
    const float hp0f = (float)headp[2 * b];
    const float hp1f = (float)headp[2 * b + 1];
    const float lenxy = sqrtf(gx * gx + gy * gy);
    const float lo1 = pd - fr,        hi1 = pd + fr;
    const float lo2 = pd - 2.0f * fr, hi2 = pd + 2.0f * fr;
    const float lo3 = pd - 3.0f * fr, hi3 = pd + 3.0f * fr;

    const int base4 = blockIdx.x * (K2_CHUNKS * THREADS);

    async_ld_b128(&sd[t], &dg[base4 + t]);
    async_ld_b128(&so[t], &og[base4 + t]);

    for (int k = 0; k < K2_CHUNKS; ++k) {
        const int cb = (k & 1) * THREADS;
        if (k + 1 < K2_CHUNKS) {
            const int nb = ((k + 1) & 1) * THREADS;
            const int g  = base4 + (k + 1) * THREADS + t;
            async_ld_b128(&sd[nb + t], &dg[g]);
            async_ld_b128(&so[nb + t], &og[g]);
            wait_async_le2();
        } else {
            wait_async_0();
        }
        float4 dv = sd[cb + t];
        float4 ov = so[cb + t];
        float dp[4] = { dv.x * ov.x, dv.y * ov.y, dv.z * ov.z, dv.w * ov.w };

        const int g4 = base4 + k * THREADS + t;   // float4 index in image
        const int p  = g4 * 4;                    // pixel index
        const int h  = p / WW;
        const int w  = p - h * WW;                // 4 pixels share row h
        const float a1 = (float)h - hp1f;

        float r0[4], r1[4], r2[4];
        #pragma unroll
        for (int i = 0; i < 4; ++i) {
            const float a0    = (float)(w + i) - hp0f;
            const float dot   = fmaf(a0, gx, a1 * gy);
            const float denom = sqrtf(fmaf(a0, a0, a1 * a1)) * lenxy;
            const float ang   = acosf(dot / denom);           // NaN if |x|>1 or 0/0
            // 12/pi = 3.8197186342054885; v_max_num drops the NaN operand,
            // exactly matching nan_to_num(maximum(..., 0)).
            const float mask = fmaxf(fmaf(-3.8197186342054885f, ang, 1.0f), 0.0f);
            const float di = dp[i];
            const float dm = di * mask;
            r0[i] = (di >= lo1 && di <= hi1) ? dm : 0.0f;
            r1[i] = (di >= lo2 && di <= hi2) ? dm : 0.0f;
            r2[i] = (di >= lo3 && di <= hi3) ? dm : 0.0f;
        }
        *(float4*)(out_b + p)          = make_float4(r0[0], r0[1], r0[2], r0[3]);
        *(float4*)(out_b + HW + p)     = make_float4(r1[0], r1[1], r1[2], r1[3]);
        *(float4*)(out_b + 2 * HW + p) = make_float4(r2[0], r2[1], r2[2], r2[3]);
    }
}

extern "C" void kernel_launch(void* const* d_in, const int* in_sizes, int n_in,
                              void* d_out, int out_size, void* d_ws, size_t ws_size,
                              hipStream_t stream) {
    (void)in_sizes; (void)n_in; (void)out_size; (void)ws_size;
    const float* depth = (const float*)d_in[0];
    const float* obj   = (const float*)d_in[1];
    const float* gaze  = (const float*)d_in[2];
    const int*   headp = (const int*)d_in[3];
    float* out    = (float*)d_out;
    float* params = (float*)d_ws;   // 128 floats

    reduce_params_kernel<<<dim3(NB), dim3(THREADS), 0, stream>>>(
        depth, obj, gaze, headp, params);
    mask_band_kernel<<<dim3(K2_BLOCKS_X, NB), dim3(THREADS), 0, stream>>>(
        depth, obj, gaze, headp, params, out);
}